// VQVAE_84378927497245
// MI455X (gfx1250) — compile-verified
//
#include <hip/hip_runtime.h>
#include <hip/hip_bf16.h>

typedef __attribute__((ext_vector_type(2))) float v2f;
typedef __attribute__((ext_vector_type(8))) float v8f;

#define WMMA_F32(A, B, Cacc) \
    __builtin_amdgcn_wmma_f32_16x16x4_f32(false, (A), false, (B), (short)0, (Cacc), false, false)

static constexpr int CCH      = 64;        // channels
static constexpr int KCODES   = 4096;      // codebook size
static constexpr int NTOK     = 65536;     // B*D*H*W tokens
static constexpr int SPATIAL  = 32768;     // D*H*W
static constexpr int ZBSTRIDE = 2097152;   // C*SPATIAL (batch stride in z/out)
static constexpr int OUT_ELEMS = 4194304;  // 2*64*8*64*64
static constexpr int LROW     = 68;        // padded LDS row stride (bank-conflict-free)

// Async copy of 16 bytes global -> LDS, tracked by ASYNCcnt.
__device__ __forceinline__ void async_copy_b128(unsigned lds_byte_addr, const float* g) {
    asm volatile("global_load_async_to_lds_b128 %0, %1, off"
                 :: "v"(lds_byte_addr), "v"(g)
                 : "memory");
}
#define WAIT_ASYNC(n) asm volatile("s_wait_asynccnt " #n ::: "memory")

// ---------------- kernel 1: half squared norms of codebook rows ----------------
__global__ void vq_ee_kernel(const float* __restrict__ emb, float* __restrict__ ee) {
    int k = blockIdx.x * blockDim.x + threadIdx.x;
    if (k >= KCODES) return;
    const float* row = emb + k * CCH;
    float s = 0.f;
    #pragma unroll
    for (int c = 0; c < CCH; ++c) s += row[c] * row[c];
    ee[k] = 0.5f * s;
}

// ---------------- kernel 2: pre conv (z bcdhw -> t token-major) via f32 WMMA ----------------
__global__ void vq_preconv_kernel(const float* __restrict__ z,
                                  const float* __restrict__ w,
                                  const float* __restrict__ bias,
                                  float* __restrict__ t) {
    const int lane = threadIdx.x & 31;
    const int lo = lane & 15, hi = lane >> 4;
    const int wid = blockIdx.x * (blockDim.x >> 5) + (threadIdx.x >> 5);
    const int tok0 = wid * 16;
    const int batch = tok0 >> 15;            // NTOK per batch = 32768
    const int sp0 = tok0 & (SPATIAL - 1);
    const float* zb = z + batch * ZBSTRIDE + sp0 + lo;

    v2f a[16];
    #pragma unroll
    for (int s = 0; s < 16; ++s) {
        int c0 = 4 * s + 2 * hi;
        a[s][0] = zb[c0 * SPATIAL];
        a[s][1] = zb[(c0 + 1) * SPATIAL];
    }
    #pragma unroll
    for (int oc = 0; oc < 4; ++oc) {
        const float* wr = w + (oc * 16 + lo) * CCH + 2 * hi;
        v8f acc0 = {}, acc1 = {};
        #pragma unroll
        for (int s = 0; s < 8; ++s) {
            v2f b0 = { wr[4 * s],        wr[4 * s + 1] };
            v2f b1 = { wr[4 * (s + 8)],  wr[4 * (s + 8) + 1] };
            acc0 = WMMA_F32(a[s],     b0, acc0);
            acc1 = WMMA_F32(a[s + 8], b1, acc1);
        }
        const float bv = bias[oc * 16 + lo];
        #pragma unroll
        for (int v = 0; v < 8; ++v)
            t[(tok0 + 8 * hi + v) * CCH + oc * 16 + lo] = acc0[v] + acc1[v] + bv;
    }
}

// ---------------- kernel 3: nearest-code search, argmax(t.e - 0.5||e||^2) ----------------
// 8 waves/WG share each 16-code embedding chunk via double-buffered async LDS staging.
__global__ void vq_dist_kernel(const float* __restrict__ t,
                               const float* __restrict__ emb,
                               const float* __restrict__ ee,
                               int* __restrict__ idx_out) {
    __shared__ float ebuf[2][16 * LROW];     // padded: bank-conflict-free ds_load_b64

    const int tid = threadIdx.x;
    const int lane = tid & 31;
    const int lo = lane & 15, hi = lane >> 4;
    const int wid = blockIdx.x * (blockDim.x >> 5) + (tid >> 5);
    const int tok0 = wid * 16;

    // A fragments: this wave's 16 tokens (resident for the whole search)
    const float* tr = t + (tok0 + lo) * CCH + 2 * hi;
    v2f a[16];
    #pragma unroll
    for (int s = 0; s < 16; ++s) { a[s][0] = tr[4 * s]; a[s][1] = tr[4 * s + 1]; }

    // async staging assignment: thread tid copies 16B: row = tid>>4, seg = (tid&15)*4 floats
    const int r = tid >> 4;
    const int seg = (tid & 15) * 4;
    const unsigned lds0 = (unsigned)(uintptr_t)&ebuf[0][r * LROW + seg];
    const unsigned lds1 = (unsigned)(uintptr_t)&ebuf[1][r * LROW + seg];
    const float* gsrc = emb + r * CCH + seg;

    float best[8];
    int bidx[8];
    #pragma unroll
    for (int v = 0; v < 8; ++v) { best[v] = -__builtin_inff(); bidx[v] = 0; }

    async_copy_b128(lds0, gsrc);             // stage chunk 0

    for (int chunk = 0; chunk < KCODES / 16; ++chunk) {
        const int cur = chunk & 1;
        if (chunk + 1 < KCODES / 16) {
            async_copy_b128(cur ? lds0 : lds1, gsrc + (chunk + 1) * 16 * CCH);
            WAIT_ASYNC(0x1);                 // our copy of 'cur' chunk is done
        } else {
            WAIT_ASYNC(0x0);
        }
        __syncthreads();                     // all waves' copies of 'cur' visible

        const float* er = &ebuf[cur][lo * LROW + 2 * hi];
        v8f acc0 = {}, acc1 = {};
        #pragma unroll
        for (int s = 0; s < 8; ++s) {
            v2f b0 = { er[4 * s],       er[4 * s + 1] };
            v2f b1 = { er[4 * (s + 8)], er[4 * (s + 8) + 1] };
            acc0 = WMMA_F32(a[s],     b0, acc0);
            acc1 = WMMA_F32(a[s + 8], b1, acc1);
        }
        const int k0 = chunk * 16;
        const float he = ee[k0 + lo];
        const int kidx = k0 + lo;
        #pragma unroll
        for (int v = 0; v < 8; ++v) {
            float sc = acc0[v] + acc1[v] - he;
            if (sc > best[v]) { best[v] = sc; bidx[v] = kidx; }
        }
        __syncthreads();                     // done reading 'cur' before it is overwritten
    }

    // reduce across the 16 lanes that share a token (xor offsets stay inside half)
    #pragma unroll
    for (int v = 0; v < 8; ++v) {
        float s = best[v];
        int bi = bidx[v];
        #pragma unroll
        for (int off = 8; off >= 1; off >>= 1) {
            float s2 = __shfl_xor(s, off, 32);
            int i2 = __shfl_xor(bi, off, 32);
            if (s2 > s || (s2 == s && i2 < bi)) { s = s2; bi = i2; }
        }
        if (lo == 0) idx_out[tok0 + 8 * hi + v] = bi;
    }
}

// ---------------- kernel 4: loss partials + index map (as float) ----------------
__global__ void vq_loss_idx_kernel(const float* __restrict__ t,
                                   const float* __restrict__ emb,
                                   const int* __restrict__ idx,
                                   float* __restrict__ idx_f_out,
                                   float* __restrict__ partial) {
    __shared__ float red[256];
    const int n = blockIdx.x * blockDim.x + threadIdx.x;
    const int id = idx[n];
    const float* q = emb + id * CCH;
    const float* tr = t + n * CCH;
    float s = 0.f;
    #pragma unroll
    for (int c = 0; c < CCH; ++c) { float d = q[c] - tr[c]; s += d * d; }
    idx_f_out[n] = (float)id;
    red[threadIdx.x] = s;
    __syncthreads();
    for (int stride = 128; stride >= 1; stride >>= 1) {
        if (threadIdx.x < stride) red[threadIdx.x] += red[threadIdx.x + stride];
        __syncthreads();
    }
    if (threadIdx.x == 0) partial[blockIdx.x] = red[0];
}

// ---------------- kernel 5: post conv (gathered codes -> out bcdhw) via f32 WMMA ----------------
__global__ void vq_postconv_kernel(const float* __restrict__ emb,
                                   const int* __restrict__ idx,
                                   const float* __restrict__ w,
                                   const float* __restrict__ bias,
                                   float* __restrict__ out) {
    const int lane = threadIdx.x & 31;
    const int lo = lane & 15, hi = lane >> 4;
    const int wid = blockIdx.x * (blockDim.x >> 5) + (threadIdx.x >> 5);
    const int tok0 = wid * 16;
    const int batch = tok0 >> 15;
    const int sp0 = tok0 & (SPATIAL - 1);

    const int id = idx[tok0 + lo];
    const float* qr = emb + id * CCH + 2 * hi;
    v2f a[16];
    #pragma unroll
    for (int s = 0; s < 16; ++s) { a[s][0] = qr[4 * s]; a[s][1] = qr[4 * s + 1]; }

    #pragma unroll
    for (int oc = 0; oc < 4; ++oc) {
        const float* wr = w + (oc * 16 + lo) * CCH + 2 * hi;
        v8f acc0 = {}, acc1 = {};
        #pragma unroll
        for (int s = 0; s < 8; ++s) {
            v2f b0 = { wr[4 * s],        wr[4 * s + 1] };
            v2f b1 = { wr[4 * (s + 8)],  wr[4 * (s + 8) + 1] };
            acc0 = WMMA_F32(a[s],     b0, acc0);
            acc1 = WMMA_F32(a[s + 8], b1, acc1);
        }
        const float bv = bias[oc * 16 + lo];
        // lane owns channel o = oc*16+lo, tokens 8*hi .. 8*hi+7 (consecutive spatial)
        float* dst = out + batch * ZBSTRIDE + (oc * 16 + lo) * SPATIAL + sp0 + 8 * hi;
        float4 p0 = { acc0[0] + acc1[0] + bv, acc0[1] + acc1[1] + bv,
                      acc0[2] + acc1[2] + bv, acc0[3] + acc1[3] + bv };
        float4 p1 = { acc0[4] + acc1[4] + bv, acc0[5] + acc1[5] + bv,
                      acc0[6] + acc1[6] + bv, acc0[7] + acc1[7] + bv };
        *(float4*)(dst)     = p0;
        *(float4*)(dst + 4) = p1;
    }
}

// ---------------- kernel 6: finalize losses (deterministic) ----------------
__global__ void vq_finalize_kernel(const float* __restrict__ partial, float* __restrict__ loss_out) {
    __shared__ float red[256];
    red[threadIdx.x] = partial[threadIdx.x];
    __syncthreads();
    for (int stride = 128; stride >= 1; stride >>= 1) {
        if (threadIdx.x < stride) red[threadIdx.x] += red[threadIdx.x + stride];
        __syncthreads();
    }
    if (threadIdx.x == 0) {
        float m = red[0] * (1.0f / (float)(NTOK * CCH));
        loss_out[0] = m;   // codebook_loss  (numerically identical)
        loss_out[1] = m;   // commitment_loss
    }
}

extern "C" void kernel_launch(void* const* d_in, const int* in_sizes, int n_in,
                              void* d_out, int out_size, void* d_ws, size_t ws_size,
                              hipStream_t stream) {
    const float* z      = (const float*)d_in[0];
    const float* emb    = (const float*)d_in[1];
    const float* pre_w  = (const float*)d_in[2];
    const float* pre_b  = (const float*)d_in[3];
    const float* post_w = (const float*)d_in[4];
    const float* post_b = (const float*)d_in[5];

    float* out      = (float*)d_out;
    float* loss_out = out + OUT_ELEMS;       // [codebook, commitment]
    float* idx_f    = out + OUT_ELEMS + 2;   // 65536 indices as float

    float* t_ws       = (float*)d_ws;                 // 65536*64 floats
    float* ee_ws      = t_ws + NTOK * CCH;            // 4096 floats
    float* partial_ws = ee_ws + KCODES;               // 256 floats
    int*   idx_ws     = (int*)(partial_ws + 256);     // 65536 ints

    vq_ee_kernel      <<<KCODES / 256, 256, 0, stream>>>(emb, ee_ws);
    vq_preconv_kernel <<<NTOK / 16 / 8, 256, 0, stream>>>(z, pre_w, pre_b, t_ws);
    vq_dist_kernel    <<<NTOK / 16 / 8, 256, 0, stream>>>(t_ws, emb, ee_ws, idx_ws);
    vq_loss_idx_kernel<<<NTOK / 256,    256, 0, stream>>>(t_ws, emb, idx_ws, idx_f, partial_ws);
    vq_postconv_kernel<<<NTOK / 16 / 8, 256, 0, stream>>>(emb, idx_ws, post_w, post_b, out);
    vq_finalize_kernel<<<1,             256, 0, stream>>>(partial_ws, loss_out);
}